// SymmetryFunctions_15023795601943
// MI455X (gfx1250) — compile-verified
//
#include <hip/hip_runtime.h>

// wACSF symmetry functions for MI455X (gfx1250, wave32).
// B=4, A=128, N=127, T=8001, 22 radial + 10 angular features per atom.
//
// Pipeline:
//   pairs_kernel   : per (b,i,n) pair record {r, w=fc*zw, ang gauss[5], w*mask, shifted pos}
//                    -> 3 MB scratch in d_ws (L2-resident).
//   angular_kernel : 1 block (256 thr) per atom.
//                    - stages the atom's 127-record pair table (6 KB) into LDS via
//                      async global->LDS copies (ASYNCcnt) when available,
//                    - threads 0..21 compute the 22 radial features from the LDS table,
//                    - all threads accumulate angular triples from the LDS table,
//                    - cross-lane reduction via v_wmma_f32_16x16x4_f32 (ones-matrix
//                      K-reduction, exact fp32), cross-wave combine via LDS slots.

#define BATCH 4
#define NA    128
#define NN    127
#define NT    8001
#define NRAD  22
#define NANG  5
#define CUTR  5.0f
#define EPSF  1e-12f
#define PIF   3.14159265358979f

typedef __attribute__((ext_vector_type(2))) float v2f;
typedef __attribute__((ext_vector_type(8))) float v8f;

// Exact parameter types for the async global->LDS builtin, per hipcc diagnostic:
// arg0: int __vector_size__(16) in AS(1) (global), non-const.
typedef int v4i_vs __attribute__((__vector_size__(16)));
typedef __attribute__((address_space(1))) v4i_vs* gv4_ptr;
typedef __attribute__((address_space(3))) v4i_vs* lv4_ptr;

#if defined(__AMDGCN__) && __has_builtin(__builtin_amdgcn_global_load_async_to_lds_b128)
#define HAVE_ASYNC_LDS 1
#else
#define HAVE_ASYNC_LDS 0
#endif

// ---------------- Stage 1: per-pair records (12 floats / pair) ----------------
// layout: f4[0] = (r, w, a0, a1)   f4[1] = (a2, a3, a4, w*mask)   f4[2] = (sx, sy, sz, 0)
__global__ __launch_bounds__(256)
void pairs_kernel(const float* __restrict__ pos,
                  const int* __restrict__ Z,
                  const int* __restrict__ nbrs,
                  const unsigned char* __restrict__ nbmask,
                  const float* __restrict__ cell,
                  const float* __restrict__ cell_off,
                  const float* __restrict__ zweight,
                  float* __restrict__ recs) {
    int tid = blockIdx.x * blockDim.x + threadIdx.x;
    if (tid >= BATCH * NA * NN) return;
    int ia = (tid / NN) % NA;
    int b  = tid / (NA * NN);
    int j  = nbrs[tid];
    const float* cb = cell + b * 9;
    const float* co = cell_off + (size_t)tid * 3;
    float ox = co[0]*cb[0] + co[1]*cb[3] + co[2]*cb[6];
    float oy = co[0]*cb[1] + co[1]*cb[4] + co[2]*cb[7];
    float oz = co[0]*cb[2] + co[1]*cb[5] + co[2]*cb[8];
    const float* pj = pos + ((size_t)b * NA + j)  * 3;
    const float* pi = pos + ((size_t)b * NA + ia) * 3;
    float sx = pj[0] + ox, sy = pj[1] + oy, sz = pj[2] + oz;
    float dx = sx - pi[0], dy = sy - pi[1], dz = sz - pi[2];
    float r2 = fmaxf(dx*dx + dy*dy + dz*dz, EPSF);   // safe_norm clamp
    float r  = sqrtf(r2);
    float fc = (r < CUTR) ? 0.5f * (__cosf(PIF * r * (1.0f / CUTR)) + 1.0f) : 0.0f;
    float zw = zweight[Z[b * NA + j]];
    float w  = fc * zw;                               // separable angular weight
    float wm = w * (nbmask[tid] ? 1.0f : 0.0f);       // radial weight incl. neighbor mask
    float a[NANG];
#pragma unroll
    for (int g = 0; g < NANG; g++) {
        float ao = 1.0f + 0.875f * (float)g;          // linspace(1.0, 4.5, 5)
        a[g] = __expf((-0.5f / (ao * ao)) * r2);      // centered angular gaussian
    }
    float4* out = (float4*)(recs + (size_t)tid * 12);
    out[0] = make_float4(r, w, a[0], a[1]);
    out[1] = make_float4(a[2], a[3], a[4], wm);
    out[2] = make_float4(sx, sy, sz, 0.0f);
}

// ---------------- Stage 2: radial + angular features (one block per atom) ----------------
__global__ __launch_bounds__(256)
void angular_kernel(const float* __restrict__ recs,
                    const int* __restrict__ oidx_j,
                    const int* __restrict__ oidx_k,
                    const unsigned char* __restrict__ tmask,
                    float* __restrict__ out) {
    __shared__ float4 ptab4[NN * 3];       // pair table: 127 records x 48 B = 6096 B
    __shared__ float  stage[8 * 32 * 16];  // per-wave 32 lanes x 16 features
    __shared__ float  wsum[8][16];         // per-wave reduced features
    const float* ptab = (const float*)ptab4;
    const int bi   = blockIdx.x;
    const int tid  = threadIdx.x;
    const int lane = tid & 31;
    const int wave = tid >> 5;

    // ---- stage this atom's pair table into LDS (async copy path if available) ----
    const float4* g4 = (const float4*)(recs + (size_t)bi * NN * 12);
#if HAVE_ASYNC_LDS
    for (int t = tid; t < NN * 3; t += 256) {
        __builtin_amdgcn_global_load_async_to_lds_b128(
            (gv4_ptr)(g4 + t), (lv4_ptr)(ptab4 + t), 0, 0);
    }
#if __has_builtin(__builtin_amdgcn_s_wait_asynccnt)
    __builtin_amdgcn_s_wait_asynccnt(0);
#else
    asm volatile("s_wait_asynccnt 0" ::: "memory");
#endif
#else
    for (int t = tid; t < NN * 3; t += 256) ptab4[t] = g4[t];
#endif
    __syncthreads();

    // ---- radial features from the LDS table (threads 0..21) ----
    if (tid < NRAD) {
        const float RSP  = 4.0f / 21.0f;          // linspace(0.5, 4.5, 22) spacing
        const float RADC = -0.5f / (RSP * RSP);
        float mu   = 0.5f + RSP * (float)tid;
        float racc = 0.0f;
        for (int n = 0; n < NN; n++) {
            float r  = ptab[n * 12 + 0];
            float wm = ptab[n * 12 + 7];
            float d  = r - mu;
            racc += __expf(RADC * d * d) * wm;
        }
        out[(size_t)bi * 32 + tid] = racc;
    }

    // ---- angular triples: per-lane accumulation from the LDS table ----
    const size_t tbase = (size_t)bi * NT;
    float acc[10];
#pragma unroll
    for (int f = 0; f < 10; f++) acc[f] = 0.0f;

    for (int t = tid; t < NT; t += 256) {
        __builtin_prefetch(&oidx_j[tbase + t + 256], 0, 1);
        __builtin_prefetch(&oidx_k[tbase + t + 256], 0, 1);
        int oj = oidx_j[tbase + t];
        int ok = oidx_k[tbase + t];
        float m = tmask[tbase + t] ? 1.0f : 0.0f;
        float4 j0 = ptab4[oj * 3 + 0], j1 = ptab4[oj * 3 + 1], j2 = ptab4[oj * 3 + 2];
        float4 k0 = ptab4[ok * 3 + 0], k1 = ptab4[ok * 3 + 1], k2 = ptab4[ok * 3 + 2];
        float rij = j0.x, rik = k0.x;
        float dxx = j2.x - k2.x, dyy = j2.y - k2.y, dzz = j2.z - k2.z;
        float rjk2  = fmaxf(dxx*dxx + dyy*dyy + dzz*dzz, EPSF);
        float denom = fmaxf(2.0f * rij * rik, EPSF);
        float cosT  = m * (rij*rij + rik*rik - rjk2) / denom;  // where(mask, ., 0)
        float s  = j0.y * k0.y * m;        // fc_j*zw_j * fc_k*zw_k * mask
        float p0 = j0.z * k0.z * s;
        float p1 = j0.w * k0.w * s;
        float p2 = j1.x * k1.x * s;
        float p3 = j1.y * k1.y * s;
        float p4 = j1.z * k1.z * s;
        float tm = 1.0f - cosT, tp = 1.0f + cosT;
        acc[0] += p0 * tm;  acc[1] += p0 * tp;
        acc[2] += p1 * tm;  acc[3] += p1 * tp;
        acc[4] += p2 * tm;  acc[5] += p2 * tp;
        acc[6] += p3 * tm;  acc[7] += p3 * tp;
        acc[8] += p4 * tm;  acc[9] += p4 * tp;
    }

    // --- cross-lane reduction via WMMA ones-matrix trick (exact fp32) ---
    // Stage P[lane][feature] (16 floats, 10 used) to LDS, then 8x
    // v_wmma_f32_16x16x4_f32 with A = P^T tile (16 feat x 4 triples), B = ones:
    // D[m,n] accumulates sum over all 32 lanes of P[t][m] in every column.
    float* st = stage + wave * 512 + lane * 16;
#pragma unroll
    for (int f = 0; f < 10; f++) st[f] = acc[f];
#pragma unroll
    for (int f = 10; f < 16; f++) st[f] = 0.0f;
    __syncthreads();

    const float* wb = stage + wave * 512;
    v8f c = {};
    v2f bones = {1.0f, 1.0f};
#pragma unroll
    for (int w8 = 0; w8 < 8; w8++) {
        // A layout (f32 16x4): lanes 0-15 hold A[M=lane, K=0..1], lanes 16-31 K=2..3
        int trip = 4 * w8 + ((lane >> 4) << 1);
        int fidx = lane & 15;
        v2f aop;
        aop.x = wb[trip * 16 + fidx];
        aop.y = wb[(trip + 1) * 16 + fidx];
        c = __builtin_amdgcn_wmma_f32_16x16x4_f32(
                /*neg_a=*/false, aop, /*neg_b=*/false, bones,
                /*c_mod=*/(short)0, c, /*reuse_a=*/false, /*reuse_b=*/false);
    }
    // D readout (16x16 f32 C/D): lane 0 VGPR v = D[v, 0]; lane 16 VGPR v = D[8+v, 0]
    if (lane == 0) {
#pragma unroll
        for (int v = 0; v < 8; v++) wsum[wave][v] = c[v];
    } else if (lane == 16) {
#pragma unroll
        for (int v = 0; v < 8; v++) wsum[wave][8 + v] = c[v];
    }
    __syncthreads();
    if (tid < 10) {
        float s = 0.0f;
#pragma unroll
        for (int w = 0; w < 8; w++) s += wsum[w][tid];
        out[(size_t)bi * 32 + NRAD + tid] = s;
    }
}

// ---------------- host launcher ----------------
extern "C" void kernel_launch(void* const* d_in, const int* in_sizes, int n_in,
                              void* d_out, int out_size, void* d_ws, size_t ws_size,
                              hipStream_t stream) {
    const float*         pos      = (const float*)d_in[0];          // B A 3
    const int*           Z        = (const int*)d_in[1];            // B A
    const int*           nbrs    = (const int*)d_in[2];             // B A N
    const unsigned char* nbmask   = (const unsigned char*)d_in[3];  // B A N (bool)
    const float*         cell     = (const float*)d_in[4];          // B 3 3
    const float*         cell_off = (const float*)d_in[5];          // B A N 3
    // d_in[6]=idx_j, d_in[7]=idx_k: not needed — idx_j == neighbors[b,i,offset_idx_j]
    // by construction of this neighbor list, so pair records cover them.
    const unsigned char* tmask    = (const unsigned char*)d_in[8];  // B A T (bool)
    const int*           oij      = (const int*)d_in[9];            // B A T
    const int*           oik      = (const int*)d_in[10];           // B A T
    const float*         zw       = (const float*)d_in[11];         // (MAXZ+1, 1)
    float* out  = (float*)d_out;                                     // B A 32
    float* recs = (float*)d_ws;                                      // needs ~3.1 MB

    const int npair = BATCH * NA * NN;
    pairs_kernel<<<(npair + 255) / 256, 256, 0, stream>>>(
        pos, Z, nbrs, nbmask, cell, cell_off, zw, recs);
    angular_kernel<<<BATCH * NA, 256, 0, stream>>>(recs, oij, oik, tmask, out);
}